// BackwardReasonModel_9234179687652
// MI455X (gfx1250) — compile-verified
//
#include <hip/hip_runtime.h>

typedef __attribute__((ext_vector_type(2))) float v2f;
typedef __attribute__((ext_vector_type(4))) float v4f;
typedef __attribute__((ext_vector_type(8))) float v8f;

#define BATCH            32
#define MAX_LOCAL_ENTITY 2048
#define NUM_FACT         524288
#define HIDDEN           128
#define N_SLOTS          (BATCH * MAX_LOCAL_ENTITY)   // 65536

// ---------------------------------------------------------------------------
// Kernel 1: rv[j] = relu( sum_k W[j,k] + b[j] )  via V_WMMA_F32_16X16X4_F32.
// D(16x16) = A(16x4) * B(4x16) + C, with A = W tile rows, B = all-ones,
// C seeded with b[] broadcast along N. Every column of D equals v for that
// 16-row tile; lane N=0 holders write the result.
// 8 waves (256 threads), one 16-row tile per wave, 32 chained WMMAs each.
// ---------------------------------------------------------------------------
__global__ __launch_bounds__(256) void rowsum_wmma_kernel(
    const float* __restrict__ W, const float* __restrict__ b,
    float* __restrict__ rv) {
  const int tid  = threadIdx.x;
  const int wave = tid >> 5;     // 0..7 -> row tile
  const int lane = tid & 31;
  const int half = lane >> 4;    // 0: K=k0+{0,1}; 1: K=k0+{2,3}
  const int mrow = lane & 15;    // A-matrix row within tile
  const int m0   = wave * 16;

  // C/D layout: lanes 0-15 VGPR r -> M=r; lanes 16-31 VGPR r -> M=8+r.
  v8f acc;
#pragma unroll
  for (int r = 0; r < 8; ++r) acc[r] = b[m0 + 8 * half + r];

  const v2f ones = {1.0f, 1.0f};  // B = all-ones (layout-independent)
  const float* wrow = W + (m0 + mrow) * HIDDEN + 2 * half;

#pragma unroll 4
  for (int k0 = 0; k0 < HIDDEN; k0 += 4) {
    v2f a;
    a[0] = wrow[k0 + 0];
    a[1] = wrow[k0 + 1];
    acc = __builtin_amdgcn_wmma_f32_16x16x4_f32(
        /*neg_a=*/false, a, /*neg_b=*/false, ones,
        /*c_mod=*/(short)0, acc, /*reuse_a=*/false, /*reuse_b=*/false);
  }

  if (mrow == 0) {  // lane 0 holds M=0..7 (col 0), lane 16 holds M=8..15
#pragma unroll
    for (int r = 0; r < 8; ++r) {
      int row = m0 + 8 * half + r;
      rv[row] = fmaxf(acc[r], 0.0f);
    }
  }
}

// ---------------------------------------------------------------------------
// Kernel 2: zero the 65536-entry count array (workspace is poisoned).
// ---------------------------------------------------------------------------
__global__ void zero_counts_kernel(unsigned* __restrict__ cnt) {
  int i = blockIdx.x * blockDim.x + threadIdx.x;
  if (i < N_SLOTS) cnt[i] = 0u;
}

// ---------------------------------------------------------------------------
// Kernel 3: histogram of tails + heads. 1M atomic increments into a 256 KB
// count array (stays resident in the 192 MB L2).
// ---------------------------------------------------------------------------
__global__ __launch_bounds__(256) void hist_kernel(
    const int* __restrict__ tails, const int* __restrict__ heads,
    unsigned* __restrict__ cnt) {
  int i = blockIdx.x * blockDim.x + threadIdx.x;
  if (i < NUM_FACT) {
    int t = __builtin_nontemporal_load(tails + i);
    int h = __builtin_nontemporal_load(heads + i);
    atomicAdd(&cnt[t], 1u);
    atomicAdd(&cnt[h], 1u);
  }
}

// ---------------------------------------------------------------------------
// Kernel 4: out[slot, j] = count[slot] * rv[j]. 32 MB of write-once output
// -> non-temporal 128-bit stores. Each block: 8 slots (1024 floats).
// ---------------------------------------------------------------------------
__global__ __launch_bounds__(256) void scale_out_kernel(
    const unsigned* __restrict__ cnt, const float* __restrict__ rv,
    float* __restrict__ out) {
  __shared__ float s_rv[HIDDEN];
  const int tid = threadIdx.x;
  if (tid < HIDDEN) s_rv[tid] = rv[tid];
  __syncthreads();

  const int slot = blockIdx.x * 8 + (tid >> 5);
  const int j    = (tid & 31) * 4;
  const float c  = (float)cnt[slot];

  v4f o;
  o.x = c * s_rv[j + 0];
  o.y = c * s_rv[j + 1];
  o.z = c * s_rv[j + 2];
  o.w = c * s_rv[j + 3];
  __builtin_nontemporal_store(o, (v4f*)(out + (size_t)slot * HIDDEN + j));
}

// ---------------------------------------------------------------------------
// Inputs (setup_inputs order):
//  0: local_entity (unused)  1: batch_heads  2: batch_rels (unused)
//  3: batch_tails            4: batch_ids (unused)  5: fact_ids (unused)
//  6: W [128x128] f32        7: b [128] f32
// Output: [32, 2048, 128] f32 = 8388608 floats.
// Workspace: [0, 256KB) counts u32[65536]; [256KB, +512B) rv f32[128].
// ---------------------------------------------------------------------------
extern "C" void kernel_launch(void* const* d_in, const int* in_sizes, int n_in,
                              void* d_out, int out_size, void* d_ws,
                              size_t ws_size, hipStream_t stream) {
  const int*   heads = (const int*)d_in[1];
  const int*   tails = (const int*)d_in[3];
  const float* W     = (const float*)d_in[6];
  const float* b     = (const float*)d_in[7];
  float*       out   = (float*)d_out;

  unsigned* cnt = (unsigned*)d_ws;
  float*    rv  = (float*)((char*)d_ws + (size_t)N_SLOTS * sizeof(unsigned));

  rowsum_wmma_kernel<<<1, 256, 0, stream>>>(W, b, rv);
  zero_counts_kernel<<<N_SLOTS / 256, 256, 0, stream>>>(cnt);
  hist_kernel<<<NUM_FACT / 256, 256, 0, stream>>>(tails, heads, cnt);
  scale_out_kernel<<<N_SLOTS / 8, 256, 0, stream>>>(cnt, rv, out);
}